// CMF_72541997629957
// MI455X (gfx1250) — compile-verified
//
#include <hip/hip_runtime.h>
#include <cmath>

// MI455X / gfx1250, wave32.
// fp32 matrix path: V_WMMA_F32_16X16X4_F32 (exact fp32, matches reference).
// Data movement: Tensor Data Mover (tensor_load_to_lds) global->LDS DMA,
// double-buffered, tracked with TENSORcnt.

typedef float v2f __attribute__((ext_vector_type(2)));
typedef float v8f __attribute__((ext_vector_type(8)));
typedef unsigned int uint;
typedef uint u32x4 __attribute__((ext_vector_type(4)));
typedef uint u32x8 __attribute__((ext_vector_type(8)));

#define C_DIM 768
#define H_DIM 56
#define W_DIM 56
#define HW    3136          // 56*56
#define BATCH 16
#define POUT  3364          // 58*58

#define A_STRIDE 18         // LDS row stride (floats) for A tiles (16 + 2 pad)
#define B_STRIDE 66         // LDS row stride (floats) for B tiles (64 + 2 pad)
#define A_BUF_BYTES (128 * A_STRIDE * 4)
#define B_BUF_BYTES (16 * B_STRIDE * 4)

static __device__ __forceinline__ v8f wmma_f32(v2f a, v2f b, v8f c) {
  // (neg_a, A, neg_b, B, c_mod, C, reuse_a, reuse_b)
  return __builtin_amdgcn_wmma_f32_16x16x4_f32(false, a, false, b, (short)0, c,
                                               false, false);
}

// --- TDM 2-D tile load: D# group0 built per call, group1 passed in. -------
// group0: [1:0]=count=1, [63:32]=lds_addr, [120:64]=global_addr, [127:126]=2
static __device__ __forceinline__ void tdm_load_2d(uint lds_addr,
                                                   const void* gptr,
                                                   u32x8 g1) {
  unsigned long long ga = (unsigned long long)(uintptr_t)gptr;
  u32x4 g0;
  g0.x = 0x1u;                                        // count=1, user D#
  g0.y = lds_addr;                                    // LDS byte address
  g0.z = (uint)(ga & 0xffffffffu);                    // global_addr[31:0]
  g0.w = (uint)((ga >> 32) & 0x01ffffffu) | 0x80000000u; // [56:32] | type=2
  asm volatile("tensor_load_to_lds %0, %1" ::"s"(g0), "s"(g1) : "memory");
}

// group1 for the A tile: 16x128 tile of a row-major matrix, stride ldK.
// data_size=4B(2), pad_enable, pad_interval=3 (16 dwords), pad_amount=1 (2 dw)
static __device__ __forceinline__ u32x8 tdm_desc_A(uint ldK) {
  u32x8 g1;
  g1[0] = (2u << 16) | (1u << 20) | (3u << 22) | (1u << 25);
  g1[1] = 16u << 16;        // tensor_dim0 = 16   ([79:48] low half)
  g1[2] = 128u << 16;       // tensor_dim1 = 128  ([111:80] low half)
  g1[3] = 16u << 16;        // tile_dim0 = 16     ([127:112])
  g1[4] = 128u;             // tile_dim1 = 128    ([143:128])
  g1[5] = ldK;              // tensor_dim0_stride[31:0]
  g1[6] = 0u;
  g1[7] = 0u;
  return g1;
}

// group1 for the B tile: 64x16 tile, row stride HW.
// pad_interval=5 (64 dwords), pad_amount=1 (2 dwords) -> LDS stride 66.
static __device__ __forceinline__ u32x8 tdm_desc_B() {
  u32x8 g1;
  g1[0] = (2u << 16) | (1u << 20) | (5u << 22) | (1u << 25);
  g1[1] = 64u << 16;        // tensor_dim0 = 64
  g1[2] = 16u << 16;        // tensor_dim1 = 16
  g1[3] = 64u << 16;        // tile_dim0 = 64
  g1[4] = 16u;              // tile_dim1 = 16
  g1[5] = (uint)HW;         // tensor_dim0_stride
  g1[6] = 0u;
  g1[7] = 0u;
  return g1;
}

// ---------------------------------------------------------------------------
// Shared GEMM core: Out(128x64 tile) = A(128xK) * B(Kx64), K = 768.
// 128 threads = 4 waves, wave grid 2(M)x2(N), each wave 64x32 = 4x2 WMMA
// tiles.  TDM double-buffered staging; wave 0 drives the DMA.
// Acc is left in `acc`, epilogue is kernel-specific.
// ---------------------------------------------------------------------------
#define GEMM_CORE(Ag, Bg)                                                     \
  __shared__ float As[2][128][A_STRIDE];                                      \
  __shared__ float Bs[2][16][B_STRIDE];                                       \
  const int t      = threadIdx.x;                                             \
  const int lane   = t & 31;                                                  \
  const int wv     = t >> 5;                                                  \
  const int wx     = wv & 1;  /* N: 2 x 32 cols  */                           \
  const int wy     = wv >> 1; /* M: 2 x 64 rows  */                           \
  const int lane16 = lane & 15;                                               \
  const int ksel   = lane >> 4;                                               \
  const uint ldsA  = (uint)(uintptr_t)&As[0][0][0];                           \
  const uint ldsB  = (uint)(uintptr_t)&Bs[0][0][0];                           \
  const u32x8 dA   = tdm_desc_A((uint)C_DIM);                                 \
  const u32x8 dB   = tdm_desc_B();                                            \
  v8f acc[4][2];                                                              \
  _Pragma("unroll") for (int i = 0; i < 4; ++i)                               \
      _Pragma("unroll") for (int j = 0; j < 2; ++j)                           \
          acc[i][j] = (v8f){0.f, 0.f, 0.f, 0.f, 0.f, 0.f, 0.f, 0.f};          \
  if (t < 32) { /* wave 0: prologue DMA into buffer 0 */                      \
    tdm_load_2d(ldsA, Ag, dA);                                                \
    tdm_load_2d(ldsB, Bg, dB);                                                \
    __builtin_amdgcn_s_wait_tensorcnt(0);                                     \
  }                                                                           \
  __syncthreads();                                                            \
  for (int kt = 0; kt < C_DIM; kt += 16) {                                    \
    const int p = (kt >> 4) & 1;                                              \
    if (t < 32 && kt + 16 < C_DIM) { /* DMA next K-tile into buffer p^1 */    \
      tdm_load_2d(ldsA + (uint)((p ^ 1) * A_BUF_BYTES), Ag + kt + 16, dA);    \
      tdm_load_2d(ldsB + (uint)((p ^ 1) * B_BUF_BYTES),                       \
                  Bg + (size_t)(kt + 16) * HW, dB);                           \
    }                                                                         \
    _Pragma("unroll") for (int kk = 0; kk < 16; kk += 4) {                    \
      const int k0 = kk + ksel * 2;                                           \
      v2f a[4], bf[2];                                                        \
      _Pragma("unroll") for (int i = 0; i < 4; ++i)                           \
          a[i] = *(const v2f*)&As[p][wy * 64 + i * 16 + lane16][k0];          \
      _Pragma("unroll") for (int j = 0; j < 2; ++j) {                         \
        const int n = wx * 32 + j * 16 + lane16;                              \
        bf[j] = (v2f){Bs[p][k0][n], Bs[p][k0 + 1][n]};                        \
      }                                                                       \
      _Pragma("unroll") for (int i = 0; i < 4; ++i)                           \
          _Pragma("unroll") for (int j = 0; j < 2; ++j)                       \
              acc[i][j] = wmma_f32(a[i], bf[j], acc[i][j]);                   \
    }                                                                         \
    if (t < 32) __builtin_amdgcn_s_wait_tensorcnt(0);                         \
    __syncthreads();                                                          \
  }

// ---------------------------------------------------------------------------
// Kernel 1: q/k/v projections.  Out[o,n] = sum_c W[o,c] * X[c,n] per batch.
// Grid: (N/64=49, M/128=6, 3*B), 128 threads.
// ---------------------------------------------------------------------------
__global__ __launch_bounds__(128)
void CMF_gemm_qkv(const float* __restrict__ x_s, const float* __restrict__ x_fq,
                  const float* __restrict__ Wq, const float* __restrict__ Wk,
                  const float* __restrict__ Wv,
                  float* __restrict__ qb, float* __restrict__ kb,
                  float* __restrict__ vb)
{
  const int which = blockIdx.z % 3;
  const int b     = blockIdx.z / 3;
  const float* Wmat = (which == 0) ? Wq : (which == 1) ? Wk : Wv;
  const float* X    = (which == 0) ? x_s : x_fq;
  float*       Out  = (which == 0) ? qb : (which == 1) ? kb : vb;

  const int Nbase = blockIdx.x * 64;
  const int Mbase = blockIdx.y * 128;
  const float* Ag = Wmat + (size_t)Mbase * C_DIM;
  const float* Bg = X + (size_t)b * C_DIM * HW + Nbase;
  float*       Cg = Out + (size_t)b * C_DIM * HW + (size_t)Mbase * HW + Nbase;

  GEMM_CORE(Ag, Bg)

  #pragma unroll
  for (int i = 0; i < 4; ++i)
    #pragma unroll
    for (int j = 0; j < 2; ++j)
      #pragma unroll
      for (int r = 0; r < 8; ++r) {
        int m = wy * 64 + i * 16 + r + 8 * ksel;
        int n = wx * 32 + j * 16 + lane16;
        Cg[(size_t)m * HW + n] = acc[i][j][r];
      }
}

// ---------------------------------------------------------------------------
// Kernel 2: per-(b,c) row attention + residual fuse (unchanged from round 1).
// ---------------------------------------------------------------------------
__global__ __launch_bounds__(128)
void CMF_attn(const float* __restrict__ qb, const float* __restrict__ kb,
              const float* __restrict__ vb, const float* __restrict__ x_s,
              const float* __restrict__ x_mt, float* __restrict__ sb,
              float scale)
{
  __shared__ float Qs[64][58];   // [h][w]
  __shared__ float Ks[64][58];   // [g][w]
  __shared__ float Vs[64][58];   // [w][g]  (V transposed)
  __shared__ float Ss[64][58];   // scores / attn

  const size_t base = (size_t)blockIdx.x * HW;
  const int t      = threadIdx.x;
  const int lane   = t & 31;
  const int wv     = t >> 5;
  const int lane16 = lane & 15;
  const int ksel   = lane >> 4;

  for (int idx = t; idx < HW; idx += 128) {
    int h = idx / W_DIM, w = idx % W_DIM;
    Qs[h][w] = qb[base + idx];
    Ks[h][w] = kb[base + idx];
    Vs[w][h] = vb[base + idx];
  }
  for (int idx = t; idx < 8 * 58; idx += 128) {
    int r = 56 + idx / 58, c0 = idx % 58;
    Qs[r][c0] = 0.f; Ks[r][c0] = 0.f; Vs[r][c0] = 0.f;
  }
  __syncthreads();

  v8f acc[4];
  #pragma unroll
  for (int j = 0; j < 4; ++j) acc[j] = (v8f){0.f,0.f,0.f,0.f,0.f,0.f,0.f,0.f};
  for (int kk = 0; kk < W_DIM; kk += 4) {
    const int k0 = kk + ksel * 2;
    v2f a = *(const v2f*)&Qs[wv * 16 + lane16][k0];
    #pragma unroll
    for (int j = 0; j < 4; ++j) {
      v2f bf = *(const v2f*)&Ks[j * 16 + lane16][k0];
      acc[j] = wmma_f32(a, bf, acc[j]);
    }
  }
  #pragma unroll
  for (int j = 0; j < 4; ++j)
    #pragma unroll
    for (int r = 0; r < 8; ++r) {
      int row = wv * 16 + r + 8 * ksel;
      int col = j * 16 + lane16;
      if (row < H_DIM && col < H_DIM) Ss[row][col] = acc[j][r] * scale;
    }
  __syncthreads();

  if (t < H_DIM) {
    float mx = -3.4e38f;
    for (int g = 0; g < H_DIM; ++g) mx = fmaxf(mx, Ss[t][g]);
    float sum = 0.f;
    for (int g = 0; g < H_DIM; ++g) {
      float e = __expf(Ss[t][g] - mx);
      Ss[t][g] = e;
      sum += e;
    }
    float inv = 1.f / sum;
    for (int g = 0; g < H_DIM; ++g) Ss[t][g] *= inv;
  }
  __syncthreads();

  v8f acc2[4];
  #pragma unroll
  for (int j = 0; j < 4; ++j) acc2[j] = (v8f){0.f,0.f,0.f,0.f,0.f,0.f,0.f,0.f};
  for (int kk = 0; kk < H_DIM; kk += 4) {
    const int k0 = kk + ksel * 2;
    v2f a = *(const v2f*)&Ss[wv * 16 + lane16][k0];
    #pragma unroll
    for (int j = 0; j < 4; ++j) {
      v2f bf = *(const v2f*)&Vs[j * 16 + lane16][k0];
      acc2[j] = wmma_f32(a, bf, acc2[j]);
    }
  }
  #pragma unroll
  for (int j = 0; j < 4; ++j)
    #pragma unroll
    for (int r = 0; r < 8; ++r) {
      int row = wv * 16 + r + 8 * ksel;
      int col = j * 16 + lane16;
      if (row < H_DIM && col < W_DIM) {
        size_t gi = base + (size_t)row * W_DIM + col;
        sb[gi] = x_s[gi] + x_mt[gi] + acc2[j][r];
      }
    }
}

// ---------------------------------------------------------------------------
// Kernel 3: padded-output border = b1[o]  (228 border pixels per 58x58 image)
// ---------------------------------------------------------------------------
__global__ __launch_bounds__(256)
void CMF_border(float* __restrict__ out, const float* __restrict__ b1)
{
  int idx = blockIdx.x * blockDim.x + threadIdx.x;
  const int total = BATCH * C_DIM * 228;
  if (idx >= total) return;
  int p  = idx % 228;
  int bc = idx / 228;
  int o  = bc % C_DIM;
  int h, w;
  if (p < 58)       { h = 0;            w = p;       }
  else if (p < 116) { h = 57;           w = p - 58;  }
  else if (p < 172) { h = p - 116 + 1;  w = 0;       }
  else              { h = p - 172 + 1;  w = 57;      }
  out[(size_t)bc * POUT + h * 58 + w] = b1[o];
}

// ---------------------------------------------------------------------------
// Kernel 4: out[b,o,h+1,w+1] = sum_c W1[o,c] * s[b,c,h,w] + b1[o]
// ---------------------------------------------------------------------------
__global__ __launch_bounds__(128)
void CMF_gemm_conv(const float* __restrict__ S, const float* __restrict__ W1,
                   const float* __restrict__ b1, float* __restrict__ out)
{
  const int b     = blockIdx.z;
  const int Nbase = blockIdx.x * 64;
  const int Mbase = blockIdx.y * 128;
  const float* Ag = W1 + (size_t)Mbase * C_DIM;
  const float* Bg = S + (size_t)b * C_DIM * HW + Nbase;

  GEMM_CORE(Ag, Bg)

  #pragma unroll
  for (int i = 0; i < 4; ++i)
    #pragma unroll
    for (int j = 0; j < 2; ++j)
      #pragma unroll
      for (int r = 0; r < 8; ++r) {
        int o = Mbase + wy * 64 + i * 16 + r + 8 * ksel;
        int n = Nbase + wx * 32 + j * 16 + lane16;
        int h = n / W_DIM, w = n % W_DIM;
        out[(size_t)b * C_DIM * POUT + (size_t)o * POUT
            + (size_t)(h + 1) * 58 + (w + 1)] = acc[i][j][r] + b1[o];
      }
}

// ---------------------------------------------------------------------------
extern "C" void kernel_launch(void* const* d_in, const int* in_sizes, int n_in,
                              void* d_out, int out_size, void* d_ws,
                              size_t ws_size, hipStream_t stream)
{
  const float* x_s  = (const float*)d_in[0];
  const float* x_fq = (const float*)d_in[1];
  const float* x_mt = (const float*)d_in[2];
  const float* Wq   = (const float*)d_in[3];
  const float* Wk   = (const float*)d_in[4];
  const float* Wv   = (const float*)d_in[5];
  const float* W1   = (const float*)d_in[6];
  const float* b1   = (const float*)d_in[7];
  float* out = (float*)d_out;

  const size_t S = (size_t)BATCH * C_DIM * HW;   // elements per q/k/v buffer
  float* qb = (float*)d_ws;                      // later reused as fused s
  float* kb = qb + S;
  float* vb = kb + S;

  // 1) q/k/v projections (3 GEMMs folded into grid.z)
  CMF_gemm_qkv<<<dim3(49, 6, 3 * BATCH), 128, 0, stream>>>(
      x_s, x_fq, Wq, Wk, Wv, qb, kb, vb);

  // 2) per-(b,c) attention + residual; writes s over the q buffer
  const float scale = 1.0f / sqrtf((float)(H_DIM * W_DIM * C_DIM));
  CMF_attn<<<dim3(BATCH * C_DIM), 128, 0, stream>>>(
      qb, kb, vb, x_s, x_mt, qb, scale);

  // 3) padded border = b1
  {
    const int total = BATCH * C_DIM * 228;
    CMF_border<<<(total + 255) / 256, 256, 0, stream>>>(out, b1);
  }

  // 4) final 1x1 conv into the interior + bias
  CMF_gemm_conv<<<dim3(49, 6, BATCH), 128, 0, stream>>>(qb, W1, b1, out);
}